// MaxPoolLayer_60473139527695
// MI455X (gfx1250) — compile-verified
//
#include <hip/hip_runtime.h>
#include <math.h>

#define D_FEAT 128

// Native CDNA5 fire-and-forget float max atomic at device scope.
// No-return form -> tracked on STOREcnt, deeply pipelined, no result bus.
__device__ __forceinline__ void atomic_fmax_dev(float* addr, float v) {
    asm volatile("global_atomic_max_num_f32 %0, %1, off scope:SCOPE_DEV"
                 :
                 : "v"(addr), "v"(v)
                 : "memory");
}

// ---------------------------------------------------------------------------
// Scatter kernel (first in file so the disasm snippet shows it):
// one wave32 per edge.
//   - edge id forced into an SGPR -> src/dst loads become scalar (s_load_b32,
//     constant cache / KMcnt), keeping the vector pipe free
//   - lane L loads float4 of node_feats[s][4L..4L+3]  (global_load_b128)
//   - lane L issues 4x global_atomic_max_num_f32 into out[d][4L..4L+3]
//   - lane 0 counts in-degree with a no-return integer add atomic
// ---------------------------------------------------------------------------
__global__ void mp_scatter_kernel(const float* __restrict__ feats,
                                  const int* __restrict__ src,
                                  const int* __restrict__ dst,
                                  float* __restrict__ out,
                                  int* __restrict__ deg,
                                  int n_edges) {
    int wave = (blockIdx.x * blockDim.x + threadIdx.x) >> 5;  // wave32
    int lane = threadIdx.x & 31;

    // wave id is wave-uniform; pin it to an SGPR so index loads scalarize.
    int e = __builtin_amdgcn_readfirstlane(wave);
    if (e >= n_edges) return;

    int s = __builtin_amdgcn_readfirstlane(src[e]);
    int d = __builtin_amdgcn_readfirstlane(dst[e]);

    // Gather: 128 f32 per edge = 32 lanes x 16B -> fully coalesced b128.
    // node_feats (5.12 MB) is L2-resident (192 MB L2), never touches HBM.
    const float4* fp = reinterpret_cast<const float4*>(
                           feats + (size_t)s * D_FEAT) + lane;
    float4 m = *fp;

    if (lane == 0) {
        atomicAdd(deg + d, 1);  // no-return global_atomic_add_u32
    }

    float* op = out + (size_t)d * D_FEAT + lane * 4;
    atomic_fmax_dev(op + 0, m.x);
    atomic_fmax_dev(op + 1, m.y);
    atomic_fmax_dev(op + 2, m.z);
    atomic_fmax_dev(op + 3, m.w);
}

// ---------------------------------------------------------------------------
// Init: output to -inf (max identity), zero the in-degree array.
// ---------------------------------------------------------------------------
__global__ void mp_init_kernel(float* __restrict__ out, int* __restrict__ deg,
                               int out_n, int n_nodes) {
    int i = blockIdx.x * blockDim.x + threadIdx.x;
    if (i < out_n)   out[i] = -__builtin_inff();
    if (i < n_nodes) deg[i] = 0;
}

// ---------------------------------------------------------------------------
// Fixup: nodes with zero in-degree get 0 (DGL max-reduce convention).
// ---------------------------------------------------------------------------
__global__ void mp_fixup_kernel(float* __restrict__ out,
                                const int* __restrict__ deg, int out_n) {
    int i = blockIdx.x * blockDim.x + threadIdx.x;
    if (i >= out_n) return;
    int node = i >> 7;  // D_FEAT == 128
    float v = out[i];
    out[i] = (deg[node] > 0) ? v : 0.0f;
}

extern "C" void kernel_launch(void* const* d_in, const int* in_sizes, int n_in,
                              void* d_out, int out_size, void* d_ws, size_t ws_size,
                              hipStream_t stream) {
    const float* feats = (const float*)d_in[0];
    const int*   src   = (const int*)d_in[1];
    const int*   dst   = (const int*)d_in[2];
    float*       out   = (float*)d_out;
    int*         deg   = (int*)d_ws;          // n_nodes ints of scratch

    const int n_edges = in_sizes[1];
    const int n_nodes = out_size / D_FEAT;

    const int threads = 256;                  // 8 wave32 per block
    const int init_blocks = (out_size + threads - 1) / threads;

    mp_init_kernel<<<init_blocks, threads, 0, stream>>>(out, deg, out_size, n_nodes);

    // one wave per edge
    long long total_threads = (long long)n_edges * 32;
    int scatter_blocks = (int)((total_threads + threads - 1) / threads);
    mp_scatter_kernel<<<scatter_blocks, threads, 0, stream>>>(
        feats, src, dst, out, deg, n_edges);

    mp_fixup_kernel<<<init_blocks, threads, 0, stream>>>(out, deg, out_size);
}